// SequenceContrastiveLoss_14791867367486
// MI455X (gfx1250) — compile-verified
//
#include <hip/hip_runtime.h>
#include <math.h>

typedef __attribute__((ext_vector_type(2))) float v2f;
typedef __attribute__((ext_vector_type(8))) float v8f;

#define DIM   512
#define BS    256
#define S_MOT 196
#define S_TXT 77
#define INV_TEMP (1.0f / 0.07f)
#define NORM_EPS 1e-12f

// ---------------------------------------------------------------------------
// Kernel 1: per-row L2 normalize + mean over sequence.
// One block (8 waves) per batch element. Wave w handles rows w, w+8, ...
// Each lane owns 16 dims (4 x float4, strided by 128 floats -> coalesced).
// ---------------------------------------------------------------------------
__global__ __launch_bounds__(256) void norm_mean_kernel(
    const float* __restrict__ src, float* __restrict__ dst, int S) {
  const int b    = blockIdx.x;
  const int tid  = threadIdx.x;
  const int wave = tid >> 5;
  const int lane = tid & 31;

  __shared__ float smem[DIM];
  smem[tid]       = 0.0f;
  smem[tid + 256] = 0.0f;
  __syncthreads();

  float acc[16];
#pragma unroll
  for (int i = 0; i < 16; ++i) acc[i] = 0.0f;

  const float* base = src + (size_t)b * (size_t)S * DIM;

  for (int s = wave; s < S; s += 8) {
    const float* row = base + (size_t)s * DIM;
    float4 x0 = *(const float4*)(row + lane * 4);
    float4 x1 = *(const float4*)(row + lane * 4 + 128);
    float4 x2 = *(const float4*)(row + lane * 4 + 256);
    float4 x3 = *(const float4*)(row + lane * 4 + 384);

    float ss = x0.x * x0.x + x0.y * x0.y + x0.z * x0.z + x0.w * x0.w;
    ss += x1.x * x1.x + x1.y * x1.y + x1.z * x1.z + x1.w * x1.w;
    ss += x2.x * x2.x + x2.y * x2.y + x2.z * x2.z + x2.w * x2.w;
    ss += x3.x * x3.x + x3.y * x3.y + x3.z * x3.z + x3.w * x3.w;

    // wave32 all-reduce
#pragma unroll
    for (int o = 16; o > 0; o >>= 1) ss += __shfl_xor(ss, o, 32);

    const float scale = 1.0f / fmaxf(sqrtf(ss), NORM_EPS);

    acc[0] += x0.x * scale;  acc[1]  += x0.y * scale;
    acc[2] += x0.z * scale;  acc[3]  += x0.w * scale;
    acc[4] += x1.x * scale;  acc[5]  += x1.y * scale;
    acc[6] += x1.z * scale;  acc[7]  += x1.w * scale;
    acc[8] += x2.x * scale;  acc[9]  += x2.y * scale;
    acc[10] += x2.z * scale; acc[11] += x2.w * scale;
    acc[12] += x3.x * scale; acc[13] += x3.y * scale;
    acc[14] += x3.z * scale; acc[15] += x3.w * scale;
  }

  // Combine the 8 waves' partial sums in LDS.
#pragma unroll
  for (int j = 0; j < 4; ++j) {
#pragma unroll
    for (int e = 0; e < 4; ++e) {
      atomicAdd(&smem[lane * 4 + j * 128 + e], acc[j * 4 + e]);
    }
  }
  __syncthreads();

  const float invS = 1.0f / (float)S;
  dst[(size_t)b * DIM + tid]       = smem[tid] * invS;
  dst[(size_t)b * DIM + tid + 256] = smem[tid + 256] * invS;
}

// ---------------------------------------------------------------------------
// Kernel 2: logits = (m @ t^T) / TEMP via V_WMMA_F32_16X16X4_F32.
// One wave per 16x16 output tile; 256 tiles -> 32 blocks x 8 waves.
// A fragment (16x4 f32): lanes 0-15 hold M=lane, K={k,k+1}; lanes 16-31 hold
// M=lane-16, K={k+2,k+3}. B fragment (4x16, B[k][n] = t[j0+n][k]) mirrors it.
// ---------------------------------------------------------------------------
__global__ __launch_bounds__(256) void gemm_wmma_kernel(
    const float* __restrict__ m, const float* __restrict__ t,
    float* __restrict__ logits) {
  const int wave = threadIdx.x >> 5;
  const int lane = threadIdx.x & 31;
  const int tile = blockIdx.x * 8 + wave;   // 0..255
  const int i0 = (tile >> 4) * 16;          // row tile
  const int j0 = (tile & 15) * 16;          // col tile

  const int half = lane >> 4;               // 0 or 1
  const int l15  = lane & 15;

  const float* arow = m + (size_t)(i0 + l15) * DIM + 2 * half; // M=l15
  const float* brow = t + (size_t)(j0 + l15) * DIM + 2 * half; // N=l15

  v8f c = {};
#pragma unroll 8
  for (int k = 0; k < DIM; k += 4) {
    v2f a = *(const v2f*)(arow + k);
    v2f b = *(const v2f*)(brow + k);
    c = __builtin_amdgcn_wmma_f32_16x16x4_f32(
        /*neg_a=*/false, a, /*neg_b=*/false, b,
        /*c_mod=*/(short)0, c, /*reuse_a=*/false, /*reuse_b=*/false);
  }

  // C/D layout: VGPR r, lanes 0-15 -> M=r, lanes 16-31 -> M=r+8; N=l15.
#pragma unroll
  for (int r = 0; r < 8; ++r) {
    const int row = i0 + r + 8 * half;
    const int col = j0 + l15;
    logits[(size_t)row * BS + col] = c[r] * INV_TEMP;
  }
}

// ---------------------------------------------------------------------------
// Kernel 3: symmetric diagonal cross-entropy.
// One block of 256 threads; thread i does online logsumexp over row i and
// column i (column reads are coalesced across threads).
// ---------------------------------------------------------------------------
__global__ __launch_bounds__(256) void loss_kernel(
    const float* __restrict__ logits, float* __restrict__ out) {
  const int i = threadIdx.x;

  float rmax = -INFINITY, rsum = 0.0f;
  float cmax = -INFINITY, csum = 0.0f;

  for (int j = 0; j < BS; ++j) {
    const float xr = logits[(size_t)i * BS + j];
    float nm = fmaxf(rmax, xr);
    rsum = rsum * __expf(rmax - nm) + __expf(xr - nm);
    rmax = nm;

    const float xc = logits[(size_t)j * BS + i];
    nm = fmaxf(cmax, xc);
    csum = csum * __expf(cmax - nm) + __expf(xc - nm);
    cmax = nm;
  }

  const float diag = logits[(size_t)i * BS + i];
  float contrib = 0.5f * ((rmax + __logf(rsum) - diag) +
                          (cmax + __logf(csum) - diag));

  __shared__ float red[BS];
  red[i] = contrib;
  __syncthreads();
#pragma unroll
  for (int s = 128; s > 0; s >>= 1) {
    if (i < s) red[i] += red[i + s];
    __syncthreads();
  }
  if (i == 0) out[0] = red[0] * (1.0f / (float)BS);
}

// ---------------------------------------------------------------------------
extern "C" void kernel_launch(void* const* d_in, const int* in_sizes, int n_in,
                              void* d_out, int out_size, void* d_ws, size_t ws_size,
                              hipStream_t stream) {
  const float* motion = (const float*)d_in[0];  // [256,196,512] f32
  const float* text   = (const float*)d_in[1];  // [256, 77,512] f32

  float* m = (float*)d_ws;          // [256,512]
  float* t = m + (size_t)BS * DIM;  // [256,512]

  float* loss   = (float*)d_out;    // [1]
  float* logits = loss + 1;         // [256,256]

  norm_mean_kernel<<<BS, 256, 0, stream>>>(motion, m, S_MOT);
  norm_mean_kernel<<<BS, 256, 0, stream>>>(text,   t, S_TXT);
  gemm_wmma_kernel<<<32, 256, 0, stream>>>(m, t, logits);
  loss_kernel<<<1, BS, 0, stream>>>(logits, loss);
}